// Model_70961449664573
// MI455X (gfx1250) — compile-verified
//
#include <hip/hip_runtime.h>
#include <stdint.h>

#define NN 8192
#define RW 10
#define H1 100
#define H2 100
#define BB 256
#define FF 16          // padded feature dim for WMMA (10 -> 16)
#define KC 128         // K-chunk staged in LDS per iteration
#define KSPLIT 2
#define KRANGE (NN / KSPLIT)

typedef __attribute__((ext_vector_type(2))) float v2f;
typedef __attribute__((ext_vector_type(8))) float v8f;

// ---- async copy (global -> LDS), tracked by ASYNCcnt ----
#if defined(__has_builtin)
#if __has_builtin(__builtin_amdgcn_s_wait_asynccnt)
#define WAIT_ASYNC(n) __builtin_amdgcn_s_wait_asynccnt(n)
#endif
#endif
#ifndef WAIT_ASYNC
#define WAIT_ASYNC(n) asm volatile("s_wait_asynccnt %0" :: "i"(n) : "memory")
#endif

// Generic pointers to LDS carry the LDS offset in their low 32 bits
// (aperture encoding); the async instruction's VDST wants that offset.
__device__ __forceinline__ unsigned lds_off(const void* p) {
  return (unsigned)(unsigned long long)p;
}

__device__ __forceinline__ void async_ld_b128(unsigned ldsoff, const void* gaddr) {
  asm volatile("global_load_async_to_lds_b128 %0, %1, off"
               :: "v"(ldsoff), "v"(gaddr) : "memory");
}

// ---------------- phase 0: build B-operand-swizzled Xpad ----------------
// For K-step `step` (4 source rows m = 4*step + r), the 64 values are laid
// out contiguously in WMMA B-fragment order: element (lane, j) holds
// Xpad[4*step + 2*(lane>>4) + j][lane&15]  ->  flat step*64 + lane*2 + j.
// A contiguous async copy then puts each step's fragment at a constant
// LDS offset, so the hot loop reads B with one ds_load_b64 per WMMA.
__global__ __launch_bounds__(256) void pad_x(const float* __restrict__ wf,
                                             float* __restrict__ xsw) {
  int i = blockIdx.x * 256 + threadIdx.x;   // over NN*FF
  if (i < NN * FF) {
    int step = i >> 6;                 // group of 4 source rows
    int w    = i & 63;
    int lane = w >> 1;
    int j    = w & 1;
    int r    = ((lane >> 4) << 1) + j; // row-in-group per ISA K striping
    int f    = lane & 15;
    int n    = (step << 2) + r;
    xsw[i] = (f < RW) ? wf[n * RW + f] : 0.0f;
  }
}

// ---------------- phase 1: hopX = hop @ Xpad via f32 WMMA ----------------
// grid: KSPLIT * 3 * (NN/128) blocks of 256 threads (8 waves).
// Each wave: one 16-row tile of hop. Double-buffered LDS with async
// prefetch of chunk i+1 overlapped with WMMA compute of chunk i
// (ASYNCcnt completes in order -> wait <= 18 releases the older chunk).
// Two accumulators break the WMMA D->C dependency chain.
__global__ __launch_bounds__(256) void hop_gemm(const float* __restrict__ hA,
                                                const float* __restrict__ hBm,
                                                const float* __restrict__ hC,
                                                const float* __restrict__ xsw,
                                                float* __restrict__ part) {
  int blk = blockIdx.x;
  int ks   = blk / (3 * (NN / 128));
  int rem  = blk % (3 * (NN / 128));
  int hopi = rem / (NN / 128);
  int rg   = rem % (NN / 128);
  const float* H = (hopi == 0) ? hA : ((hopi == 1) ? hBm : hC);
  int row0 = rg * 128;
  int wave = threadIdx.x >> 5;
  int lane = threadIdx.x & 31;

  __shared__ float sX[2][KC / 4][64];     // 16 KB, B fragments in load order
  __shared__ float sH[2][8][16][132];     // 132 KB, per-wave hop tiles (132: bank pad)

  v8f acc0 = {};
  v8f acc1 = {};
  int kbase = ks * KRANGE;
  int rsel  = lane & 15;
  int khalf = (lane >> 4) << 1;           // 0 or 2 (K placement per ISA A layout)

  // 18 async ops per chunk per wave: 2 (sX share) + 16 (own sH tile)
  auto stage = [&](int bsel, int kc) {
    const float* gx = xsw + (size_t)kc * FF;   // contiguous, already swizzled
    float* sx0 = &sX[bsel][0][0];
    int i0 = threadIdx.x * 4;
    async_ld_b128(lds_off(sx0 + i0),        gx + i0);
    async_ld_b128(lds_off(sx0 + i0 + 1024), gx + i0 + 1024);
    const float* gh = H + (size_t)(row0 + wave * 16) * NN + kc + lane * 4;
    #pragma unroll 4
    for (int r = 0; r < 16; ++r)
      async_ld_b128(lds_off(&sH[bsel][wave][r][lane * 4]), gh + (size_t)r * NN);
  };

  const int nch = KRANGE / KC;            // 32 chunks
  stage(0, kbase);                        // prologue: chunk 0 -> buffer 0

  for (int ci = 0; ci < nch; ++ci) {
    int cur = ci & 1;
    if (ci + 1 < nch) {
      stage(cur ^ 1, kbase + (ci + 1) * KC);  // prefetch next chunk
      WAIT_ASYNC(18);                         // older 18 ops (chunk ci) done
    } else {
      WAIT_ASYNC(0);
    }
    __syncthreads();                          // chunk ci visible to all waves

    const float* sa = &sH[cur][wave][rsel][khalf]; // A frag base (b64 per step)
    const float* sb = &sX[cur][0][lane * 2];       // B frag base (b64 per step)
    #pragma unroll 4
    for (int s = 0; s < KC / 8; ++s) {
      v2f a0, a1;
      a0[0] = sa[s * 8];                    // A: VGPR0 = K0/K2, VGPR1 = K1/K3
      a0[1] = sa[s * 8 + 1];
      v2f b0 = *(const v2f*)(sb + s * 128); // step 2s, constant LDS offset
      acc0 = __builtin_amdgcn_wmma_f32_16x16x4_f32(
          false, a0, false, b0, (short)0, acc0, false, false);
      a1[0] = sa[s * 8 + 4];
      a1[1] = sa[s * 8 + 5];
      v2f b1 = *(const v2f*)(sb + s * 128 + 64); // step 2s+1
      acc1 = __builtin_amdgcn_wmma_f32_16x16x4_f32(
          false, a1, false, b1, (short)0, acc1, false, false);
    }
    __syncthreads();                          // all reads of buf `cur` done
  }

  v8f acc = acc0 + acc1;

  // D layout: VGPR r -> row r (lanes0-15) / row r+8 (lanes16-31), N = lane%16
  float* dst = part + ((size_t)(ks * 3 + hopi) * NN + row0 + wave * 16) * FF;
  int rofs = (lane >> 4) * 8;
  int col  = lane & 15;
  #pragma unroll
  for (int r = 0; r < 8; ++r)
    dst[(size_t)(rofs + r) * FF + col] = acc[r];
}

// ---------------- phase 2: 2-layer MLP per node, summed over k ----------------
__global__ __launch_bounds__(128) void mlp(const float* __restrict__ wf,
                                           const float* __restrict__ part,
                                           const float* __restrict__ W1,
                                           const float* __restrict__ b1,
                                           const float* __restrict__ W2,
                                           const float* __restrict__ b2,
                                           float* __restrict__ nodeo) {
  __shared__ float sW1[RW * H1];     // 4 KB
  __shared__ float sW2[H1 * H2];     // 40 KB
  __shared__ float sb1[H1];
  __shared__ float sb2[H2];

  int t = threadIdx.x;
  int n = blockIdx.x * 128 + t;

  float acc[H2];
  #pragma unroll
  for (int o = 0; o < H2; ++o) acc[o] = 0.0f;

  for (int k = 0; k < 4; ++k) {
    __syncthreads();
    for (int i = t; i < RW * H1; i += 128) sW1[i] = W1[k * RW * H1 + i];
    for (int i = t; i < H1 * H2; i += 128) sW2[i] = W2[k * H1 * H2 + i];
    if (t < H1) sb1[t] = b1[k * H1 + t];
    if (t < H2) sb2[t] = b2[k * H2 + t];
    __syncthreads();

    float x[RW];
    if (k == 0) {
      #pragma unroll
      for (int i = 0; i < RW; ++i) x[i] = wf[(size_t)n * RW + i];
    } else {
      const float* p0 = part + ((size_t)(0 * 3 + (k - 1)) * NN + n) * FF;
      const float* p1 = part + ((size_t)(1 * 3 + (k - 1)) * NN + n) * FF;
      #pragma unroll
      for (int i = 0; i < RW; ++i) x[i] = p0[i] + p1[i];
    }

    for (int j = 0; j < H1; ++j) {
      float h = sb1[j];
      #pragma unroll
      for (int i = 0; i < RW; ++i) h = fmaf(x[i], sW1[i * H1 + j], h);
      h = fmaxf(h, 0.0f);
      const float* w2r = &sW2[j * H2];
      #pragma unroll
      for (int o = 0; o < H2; ++o) acc[o] = fmaf(h, w2r[o], acc[o]);
    }
    #pragma unroll
    for (int o = 0; o < H2; ++o) acc[o] += sb2[o];
  }

  #pragma unroll
  for (int o = 0; o < H2; ++o) nodeo[(size_t)n * H2 + o] = acc[o];
}

// ---------------- phase 3: deterministic segment sum (batch_idx sorted) ----------------
__device__ __forceinline__ int lbound(const int* __restrict__ a, int n, int v) {
  int lo = 0, hi = n;
  while (lo < hi) { int m = (lo + hi) >> 1; if (a[m] < v) lo = m + 1; else hi = m; }
  return lo;
}

__global__ __launch_bounds__(128) void pool(const int* __restrict__ bidx,
                                            const float* __restrict__ nodeo,
                                            float* __restrict__ out) {
  int b = blockIdx.x, t = threadIdx.x;
  int lo = lbound(bidx, NN, b);
  int hi = lbound(bidx, NN, b + 1);
  if (t < H2) {
    float s = 0.0f;
    for (int n = lo; n < hi; ++n) s += nodeo[(size_t)n * H2 + t];
    out[(size_t)b * H2 + t] = s;
  }
}

// ---------------- host ----------------
extern "C" void kernel_launch(void* const* d_in, const int* in_sizes, int n_in,
                              void* d_out, int out_size, void* d_ws, size_t ws_size,
                              hipStream_t stream) {
  const float* wf   = (const float*)d_in[0];
  const float* hop1 = (const float*)d_in[1];
  const float* hop2 = (const float*)d_in[2];
  const float* hop3 = (const float*)d_in[3];
  const int*   bidx = (const int*)d_in[4];
  const float* W1   = (const float*)d_in[5];
  const float* b1   = (const float*)d_in[6];
  const float* W2   = (const float*)d_in[7];
  const float* b2   = (const float*)d_in[8];
  float* out = (float*)d_out;

  float* xsw   = (float*)d_ws;                        // NN*FF (B-swizzled)
  float* part  = xsw + (size_t)NN * FF;               // KSPLIT*3*NN*FF
  float* nodeo = part + (size_t)KSPLIT * 3 * NN * FF; // NN*H2

  pad_x<<<(NN * FF + 255) / 256, 256, 0, stream>>>(wf, xsw);
  hop_gemm<<<KSPLIT * 3 * (NN / 128), 256, 0, stream>>>(hop1, hop2, hop3, xsw, part);
  mlp<<<NN / 128, 128, 0, stream>>>(wf, part, W1, b1, W2, b2, nodeo);
  pool<<<BB, 128, 0, stream>>>(bidx, nodeo, out);
}